// COSNetModified_26508538151774
// MI455X (gfx1250) — compile-verified
//
#include <hip/hip_runtime.h>
#include <cstdint>
#include <cstddef>

// Problem constants
#define NIMG 16
#define HW   262144            // 512*512
#define WPR  16                // u32 words per 512-pixel row
#define WPI  8192              // u32 words per image bitmap (512 rows * 16)
#define NSL  8                 // slices per image for the bandwidth passes

// ---------------- CDNA5 feature detection (device pass only) ----------------
#if defined(__gfx1250__) && __has_builtin(__builtin_amdgcn_wmma_f32_16x16x4_f32)
#define USE_WMMA 1
#else
#define USE_WMMA 0
#endif

#if defined(__gfx1250__) && __has_builtin(__builtin_amdgcn_tensor_load_to_lds) && \
    __has_builtin(__builtin_amdgcn_tensor_store_from_lds) && \
    __has_builtin(__builtin_amdgcn_s_wait_tensorcnt)
#define USE_TDM 1
#else
#define USE_TDM 0
#endif

#if defined(__gfx1250__) && __has_builtin(__builtin_amdgcn_ballot_w32)
#define USE_BALLOT32 1
#else
#define USE_BALLOT32 0
#endif

#if defined(__gfx1250__) && __has_builtin(__builtin_amdgcn_rcpf)
#define USE_RCP 1
#else
#define USE_RCP 0
#endif

typedef float v2f __attribute__((ext_vector_type(2)));
typedef float v8f __attribute__((ext_vector_type(8)));

__device__ __forceinline__ float sigm(float x) {
#if USE_RCP
  // native v_exp_f32 + v_rcp_f32: avoids the IEEE divide expansion
  return __builtin_amdgcn_rcpf(1.0f + __expf(-x));
#else
  return 1.0f / (1.0f + __expf(-x));
#endif
}

// ======================= Kernel A: partial sum / sumsq / nonzero count =======
// grid (NIMG, 2, NSL): 256 blocks. WMMA f32 16x16x4 with A=ones runs the f32
// reduction on the matrix pipe (each value counted 16x -> /16 at the end).
__global__ __launch_bounds__(512)
void k_stats(const float* __restrict__ thick, const float* __restrict__ thin,
             float* __restrict__ wsPS, float* __restrict__ wsPQ,
             unsigned* __restrict__ wsPC) {
  const int n = blockIdx.x, t = blockIdx.y, sl = blockIdx.z, tid = threadIdx.x;
  const int CH4 = HW / 4 / NSL;   // float4 per slice = 8192
  const float4* src = (const float4*)((t == 0 ? thick : thin) + (size_t)n * HW) + sl * CH4;

  unsigned cnt = 0;
  float accS = 0.f, accQ = 0.f;
#if USE_WMMA
  v2f ones; ones[0] = 1.f; ones[1] = 1.f;
  v8f cs = {0.f,0.f,0.f,0.f,0.f,0.f,0.f,0.f};
  v8f cq = {0.f,0.f,0.f,0.f,0.f,0.f,0.f,0.f};
#endif

  for (int i = tid; i < CH4; i += 512) {
    float4 v = src[i];
    float s0 = sigm(v.x), s1 = sigm(v.y), s2 = sigm(v.z), s3 = sigm(v.w);
    cnt += (s0 > 0.f) + (s1 > 0.f) + (s2 > 0.f) + (s3 > 0.f);
#if USE_WMMA
    v2f b0; b0[0] = s0;      b0[1] = s1;
    v2f b1; b1[0] = s2;      b1[1] = s3;
    v2f q0; q0[0] = s0 * s0; q0[1] = s1 * s1;
    v2f q1; q1[0] = s2 * s2; q1[1] = s3 * s3;
    cs = __builtin_amdgcn_wmma_f32_16x16x4_f32(false, ones, false, b0, (short)0, cs, false, false);
    cs = __builtin_amdgcn_wmma_f32_16x16x4_f32(false, ones, false, b1, (short)0, cs, false, false);
    cq = __builtin_amdgcn_wmma_f32_16x16x4_f32(false, ones, false, q0, (short)0, cq, false, false);
    cq = __builtin_amdgcn_wmma_f32_16x16x4_f32(false, ones, false, q1, (short)0, cq, false, false);
#else
    accS += s0 + s1 + s2 + s3;
    accQ += s0 * s0 + s1 * s1 + s2 * s2 + s3 * s3;
#endif
  }
#if USE_WMMA
  accS = (cs[0]+cs[1]+cs[2]+cs[3]+cs[4]+cs[5]+cs[6]+cs[7]) * (1.0f / 16.0f);
  accQ = (cq[0]+cq[1]+cq[2]+cq[3]+cq[4]+cq[5]+cq[6]+cq[7]) * (1.0f / 16.0f);
#endif

  __shared__ float    rs[512];
  __shared__ float    rq[512];
  __shared__ unsigned rc[512];
  rs[tid] = accS; rq[tid] = accQ; rc[tid] = cnt;
  __syncthreads();
  for (int s = 256; s > 0; s >>= 1) {
    if (tid < s) { rs[tid] += rs[tid+s]; rq[tid] += rq[tid+s]; rc[tid] += rc[tid+s]; }
    __syncthreads();
  }
  if (tid == 0) {
    const int pi = (t * NIMG + n) * NSL + sl;
    wsPS[pi] = rs[0];
    wsPQ[pi] = rq[0];
    wsPC[pi] = rc[0];
  }
}

// Deterministic finalize: fixed-order sum of the NSL partials.
__global__ __launch_bounds__(32)
void k_fin_stats(const float* __restrict__ wsPS, const float* __restrict__ wsPQ,
                 const unsigned* __restrict__ wsPC,
                 float* __restrict__ wsSum, float* __restrict__ wsSq,
                 unsigned* __restrict__ wsCnt) {
  const int si = threadIdx.x;
  if (si >= 2 * NIMG) return;
  float s = 0.f, q = 0.f; unsigned c = 0;
  for (int j = 0; j < NSL; ++j) {
    s += wsPS[si * NSL + j];
    q += wsPQ[si * NSL + j];
    c += wsPC[si * NSL + j];
  }
  wsSum[si] = s; wsSq[si] = q; wsCnt[si] = c;
}

// ======================= Kernel B: "is threshold empty?" partial counts ======
__global__ __launch_bounds__(512)
void k_count(const float* __restrict__ thick, const float* __restrict__ thin,
             const float* __restrict__ wsSum, const float* __restrict__ wsSq,
             const unsigned* __restrict__ wsCnt,
             unsigned* __restrict__ wsPM, unsigned* __restrict__ wsPK) {
  const int n = blockIdx.x, t = blockIdx.y, sl = blockIdx.z, tid = threadIdx.x;
  const int si = t * NIMG + n;
  unsigned c = wsCnt[si];
  float cntf = (float)(c ? c : 1u);
  float mean = wsSum[si] / cntf;
  float var  = fmaxf(wsSq[si] / cntf - mean * mean, 0.f);
  float sd   = sqrtf(var);
  float fm   = (t == 0) ? 2.0f : 4.0f;
  float Tm = mean + fm * sd;
  float Tk = mean + 0.5f * sd;

  const int CH4 = HW / 4 / NSL;
  const float4* src = (const float4*)((t == 0 ? thick : thin) + (size_t)n * HW) + sl * CH4;
  unsigned cm = 0, ck = 0;
  for (int i = tid; i < CH4; i += 512) {
    float4 v = src[i];
    float s0 = sigm(v.x), s1 = sigm(v.y), s2 = sigm(v.z), s3 = sigm(v.w);
    cm += (s0 > Tm) + (s1 > Tm) + (s2 > Tm) + (s3 > Tm);
    ck += (s0 > Tk) + (s1 > Tk) + (s2 > Tk) + (s3 > Tk);
  }
  __shared__ unsigned rm[512];
  __shared__ unsigned rk[512];
  rm[tid] = cm; rk[tid] = ck;
  __syncthreads();
  for (int s = 256; s > 0; s >>= 1) {
    if (tid < s) { rm[tid] += rm[tid+s]; rk[tid] += rk[tid+s]; }
    __syncthreads();
  }
  if (tid == 0) {
    const int pi = si * NSL + sl;
    wsPM[pi] = rm[0]; wsPK[pi] = rk[0];
  }
}

__global__ __launch_bounds__(32)
void k_fin_count(const unsigned* __restrict__ wsPM, const unsigned* __restrict__ wsPK,
                 unsigned* __restrict__ wsCntM, unsigned* __restrict__ wsCntK) {
  const int si = threadIdx.x;
  if (si >= 2 * NIMG) return;
  unsigned m = 0, k = 0;
  for (int j = 0; j < NSL; ++j) { m += wsPM[si * NSL + j]; k += wsPK[si * NSL + j]; }
  wsCntM[si] = m; wsCntK[si] = k;
}

// ======================= Kernel C: pack marker/mask bitmaps (wave32 ballot) ==
__global__ __launch_bounds__(512)
void k_binmaps(const float* __restrict__ thick, const float* __restrict__ thin,
               const float* __restrict__ wsSum, const float* __restrict__ wsSq,
               const unsigned* __restrict__ wsCnt,
               const unsigned* __restrict__ wsCntM, const unsigned* __restrict__ wsCntK,
               unsigned* __restrict__ wsMask, unsigned* __restrict__ wsRec) {
  const int n = blockIdx.x, t = blockIdx.y, sl = blockIdx.z;
  const int si = t * NIMG + n;
  unsigned c = wsCnt[si];
  float cntf = (float)(c ? c : 1u);
  float mean = wsSum[si] / cntf;
  float var  = fmaxf(wsSq[si] / cntf - mean * mean, 0.f);
  float sd   = sqrtf(var);
  float fm   = (t == 0) ? 2.0f : 4.0f;
  float Tm = (wsCntM[si] > 0) ? (mean + fm * sd)   : (mean + 0.5f * fm * sd);
  float Tk = (wsCntK[si] > 0) ? (mean + 0.5f * sd) : (mean + 0.25f * sd);

  const float* img = (t == 0 ? thick : thin) + (size_t)n * HW;
  unsigned* mOut = wsMask + (size_t)si * WPI;
  unsigned* rOut = wsRec  + (size_t)si * WPI;

  const int lane = threadIdx.x & 31;
  const int wave = threadIdx.x >> 5;
  const int W0 = sl * (WPI / NSL);          // 1024 words per slice
  const int W1 = W0 + (WPI / NSL);
  for (int w = W0 + wave; w < W1; w += 16) {
    float s = sigm(img[w * 32 + lane]);
    bool mb = s > Tm;   // marker
    bool kb = s > Tk;   // mask
#if USE_BALLOT32
    unsigned bm = __builtin_amdgcn_ballot_w32(mb);
    unsigned bk = __builtin_amdgcn_ballot_w32(kb);
#else
    unsigned bm = (unsigned)__ballot(mb);
    unsigned bk = (unsigned)__ballot(kb);
#endif
    if (lane == 0) { rOut[w] = bm; mOut[w] = bk; }
  }
}

// ======================= Kernel D: reconstruction-by-dilation in LDS =========
#if USE_TDM
typedef unsigned int u32x4_t __attribute__((ext_vector_type(4)));
typedef int          i32x4_t __attribute__((ext_vector_type(4)));
typedef int          i32x8_t __attribute__((ext_vector_type(8)));

// 1D 32KB bitmap copy: data_size=8B, tile_dim0 = 4096 elements.
// 6-arg toolchain: (g0 u32x4, g1 i32x8, g2 i32x4, g3 i32x4, g4 i32x8, cpol).
__device__ __forceinline__ void tdm_bitmap(unsigned ldsAddr, const unsigned* gptr, bool isStore) {
  unsigned long long ga = (unsigned long long)(const void*)gptr;
  u32x4_t g0;
  g0[0] = 1u;                                              // count=1, user mode
  g0[1] = ldsAddr;                                         // lds_addr
  g0[2] = (unsigned)ga;                                    // global_addr[31:0]
  g0[3] = (unsigned)((ga >> 32) & 0x01FFFFFFu) | (2u<<30); // global_addr[56:32] | type=2
  i32x8_t g1;
  g1[0] = (int)(3u << 16);         // data_size = 8 bytes
  g1[1] = (int)(4096u << 16);      // tensor_dim0[15:0] = 4096
  g1[2] = (int)(1u << 16);         // tensor_dim0[31:16]=0 | tensor_dim1 lo16 = 1
  g1[3] = (int)(4096u << 16);      // tensor_dim1 hi16=0 | tile_dim0 = 4096
  g1[4] = 0;                       // tile_dim1 = 0 (unused), tile_dim2 = 0
  g1[5] = (int)4096u;              // tensor_dim0_stride lo32
  g1[6] = 0;
  g1[7] = 0;
  i32x4_t gz4; gz4[0]=0; gz4[1]=0; gz4[2]=0; gz4[3]=0;
  i32x8_t gz8; gz8[0]=0; gz8[1]=0; gz8[2]=0; gz8[3]=0; gz8[4]=0; gz8[5]=0; gz8[6]=0; gz8[7]=0;
  if (isStore) __builtin_amdgcn_tensor_store_from_lds(g0, g1, gz4, gz4, gz8, 0);
  else         __builtin_amdgcn_tensor_load_to_lds(g0, g1, gz4, gz4, gz8, 0);
}
#endif

__global__ __launch_bounds__(512)
void k_reconstruct(const unsigned* __restrict__ wsMask, unsigned* __restrict__ wsRec) {
  __shared__ unsigned sRec[WPI];   // 32 KB: current reconstruction bitmap
  __shared__ int sChanged;
  const int n = blockIdx.x, t = blockIdx.y, tid = threadIdx.x;
  const int si = t * NIMG + n;
  const unsigned* mG = wsMask + (size_t)si * WPI;
  unsigned*       rG = wsRec  + (size_t)si * WPI;

#if USE_TDM
  if (tid < 32) {   // wave 0 issues the DMA once; EXEC is ignored by TDM
    tdm_bitmap((unsigned)(unsigned long long)(void*)&sRec[0], rG, false);
    __builtin_amdgcn_s_wait_tensorcnt(0);
  }
#else
  for (int i = tid; i < WPI; i += 512) sRec[i] = rG[i];
#endif

  // Each thread owns one row; its mask row never changes -> keep in registers.
  unsigned m[WPR];
  {
    const uint4* mr = (const uint4*)(mG + tid * WPR);
    uint4 a = mr[0], b = mr[1], cc = mr[2], d = mr[3];
    m[0]=a.x;  m[1]=a.y;  m[2]=a.z;  m[3]=a.w;
    m[4]=b.x;  m[5]=b.y;  m[6]=b.z;  m[7]=b.w;
    m[8]=cc.x; m[9]=cc.y; m[10]=cc.z; m[11]=cc.w;
    m[12]=d.x; m[13]=d.y; m[14]=d.z; m[15]=d.w;
  }
  __syncthreads();

  const int rowBase = tid * WPR;
  for (int iter = 0; iter < 1024; ++iter) {
    if (tid == 0) sChanged = 0;
    unsigned cur[WPR], t0[WPR];
#pragma unroll
    for (int w = 0; w < WPR; ++w) {
      unsigned c0 = sRec[rowBase + w];
      unsigned up = (tid > 0)   ? sRec[rowBase - WPR + w] : 0u;
      unsigned dn = (tid < 511) ? sRec[rowBase + WPR + w] : 0u;
      cur[w] = c0;
      t0[w] = (c0 | up | dn) & m[w];
    }
    // Extra horizontal smears accelerate convergence; bounded by mask so the
    // fixed point is identical to the reference's single-step dilation.
#pragma unroll
    for (int s = 0; s < 6; ++s) {
      unsigned nx[WPR];
#pragma unroll
      for (int w = 0; w < WPR; ++w) {
        unsigned v  = t0[w];
        unsigned lo = (w > 0)       ? t0[w-1] : 0u;
        unsigned hi = (w < WPR - 1) ? t0[w+1] : 0u;
        nx[w] = (v | (v << 1) | (v >> 1) | (lo >> 31) | (hi << 31)) & m[w];
      }
#pragma unroll
      for (int w = 0; w < WPR; ++w) t0[w] = nx[w];
    }
    unsigned diff = 0;
#pragma unroll
    for (int w = 0; w < WPR; ++w) diff |= (t0[w] ^ cur[w]);

    __syncthreads();                              // reads done; reset visible
#pragma unroll
    for (int w = 0; w < WPR; ++w) sRec[rowBase + w] = t0[w];
    if (diff) sChanged = 1;
    __syncthreads();                              // writes + flags done
    int again = sChanged;
    __syncthreads();                              // flag read before next reset
    if (!again) break;
  }
  __syncthreads();

#if USE_TDM
  if (tid < 32) {
    tdm_bitmap((unsigned)(unsigned long long)(void*)&sRec[0], rG, true);
    __builtin_amdgcn_s_wait_tensorcnt(0);
  }
#else
  for (int i = tid; i < WPI; i += 512) rG[i] = sRec[i];
#endif
}

// ======================= Kernel E: passthrough copies + fused output =========
__global__ __launch_bounds__(256)
void k_outputs(const float* __restrict__ thick, const float* __restrict__ thin,
               const unsigned* __restrict__ wsRec, const unsigned* __restrict__ wsCnt,
               float* __restrict__ out) {
  const int TOT4 = NIMG * HW / 4;        // 1,048,576 float4 per tensor
  const int i = blockIdx.x * 256 + threadIdx.x;
  if (i >= TOT4) return;
  const float4* t4 = (const float4*)thick;
  const float4* n4 = (const float4*)thin;
  float4* o4 = (float4*)out;
  o4[i]         = t4[i];
  o4[i + TOT4]  = n4[i];

  int p   = i * 4;
  int n   = p >> 18;                     // / HW
  int pin = p & (HW - 1);
  int w   = pin >> 5;
  int bit = pin & 31;
  unsigned wt = wsRec[(size_t)n * WPI + w];                 // thick rec bits
  unsigned wn = wsRec[(size_t)(NIMG + n) * WPI + w];        // thin  rec bits
  if (wsCnt[n] == 0)        wt = 0;
  if (wsCnt[NIMG + n] == 0) wn = 0;
  unsigned fb = wt | wn;
  float4 f;
  f.x = ((fb >> (bit + 0)) & 1u) ? 1.0f : 0.0f;
  f.y = ((fb >> (bit + 1)) & 1u) ? 1.0f : 0.0f;
  f.z = ((fb >> (bit + 2)) & 1u) ? 1.0f : 0.0f;
  f.w = ((fb >> (bit + 3)) & 1u) ? 1.0f : 0.0f;
  o4[i + 2 * TOT4] = f;
}

// ======================= Host launch =========================================
extern "C" void kernel_launch(void* const* d_in, const int* in_sizes, int n_in,
                              void* d_out, int out_size, void* d_ws, size_t ws_size,
                              hipStream_t stream) {
  (void)in_sizes; (void)n_in; (void)out_size; (void)ws_size;
  const float* thick = (const float*)d_in[0];
  const float* thin  = (const float*)d_in[1];
  float* out = (float*)d_out;

  // Workspace layout (uses ~2.03 MB of d_ws):
  //   [0    ,  256) floats : sum[32], sumsq[32]
  //   [512  ,  896) bytes  : cnt[32], cntM[32], cntK[32] (unsigned)
  //   [1024 , 2048) floats : partial sums  [32*NSL]
  //   [2048 , 3072) floats : partial sumsq [32*NSL]
  //   [3072 , 4096) u32    : partial cnt   [32*NSL]
  //   [4096 , 5120) u32    : partial cntM  [32*NSL]
  //   [5120 , 6144) u32    : partial cntK  [32*NSL]
  //   [32768, +1MB)        : mask bitmaps [2][16][8192] u32
  //   [+1MB , +2MB)        : rec  bitmaps [2][16][8192] u32
  float*    wsSum  = (float*)d_ws;
  float*    wsSq   = wsSum + 32;
  unsigned* wsCnt  = (unsigned*)((char*)d_ws + 512);
  unsigned* wsCntM = wsCnt + 32;
  unsigned* wsCntK = wsCnt + 64;
  float*    wsPS   = (float*)((char*)d_ws + 1024);
  float*    wsPQ   = (float*)((char*)d_ws + 2048);
  unsigned* wsPC   = (unsigned*)((char*)d_ws + 3072);
  unsigned* wsPM   = (unsigned*)((char*)d_ws + 4096);
  unsigned* wsPK   = (unsigned*)((char*)d_ws + 5120);
  unsigned* wsMask = (unsigned*)((char*)d_ws + 32768);
  unsigned* wsRec  = wsMask + 32 * WPI;

  dim3 gridS(NIMG, 2, NSL);   // 256 blocks for the bandwidth passes
  dim3 gridR(NIMG, 2);        // 32 blocks for the LDS flood fill

  k_stats      <<<gridS, 512, 0, stream>>>(thick, thin, wsPS, wsPQ, wsPC);
  k_fin_stats  <<<1, 32, 0, stream>>>(wsPS, wsPQ, wsPC, wsSum, wsSq, wsCnt);
  k_count      <<<gridS, 512, 0, stream>>>(thick, thin, wsSum, wsSq, wsCnt, wsPM, wsPK);
  k_fin_count  <<<1, 32, 0, stream>>>(wsPM, wsPK, wsCntM, wsCntK);
  k_binmaps    <<<gridS, 512, 0, stream>>>(thick, thin, wsSum, wsSq, wsCnt, wsCntM, wsCntK,
                                           wsMask, wsRec);
  k_reconstruct<<<gridR, 512, 0, stream>>>(wsMask, wsRec);
  k_outputs    <<<(NIMG * HW / 4) / 256, 256, 0, stream>>>(thick, thin, wsRec, wsCnt, out);
}